// ArcFaceLoss_4286377361898
// MI455X (gfx1250) — compile-verified
//
#include <hip/hip_runtime.h>
#include <math.h>

typedef __attribute__((ext_vector_type(2))) float v2f;
typedef __attribute__((ext_vector_type(8))) float v8f;
typedef __attribute__((ext_vector_type(4))) int   v4i;
typedef __attribute__((address_space(1))) v4i*    gptr_v4i;   // global (HIP __device__)
typedef __attribute__((address_space(3))) v4i*    lptr_v4i;   // LDS (HIP __shared__)

#define B_ROWS 8192
#define D_DIM  512
#define NJB    64          // 8192 / 128 column blocks
#define LDSW   36          // padded row stride (floats): 16B-aligned, bank-conflict-free

#if __has_builtin(__builtin_amdgcn_global_load_async_to_lds_b128) && \
    __has_builtin(__builtin_amdgcn_s_wait_asynccnt)
#define USE_ASYNC_LDS 1
#else
#define USE_ASYNC_LDS 0
#endif

// ---------------------------------------------------------------------------
// Kernel 1: normalize x rows and gathered W[labels] rows into workspace (f32)
// one wave (32 lanes) per row; 16 floats/lane as 4x float4
// ---------------------------------------------------------------------------
__global__ __launch_bounds__(256) void normalize_rows(const float* __restrict__ x,
                                                      const float* __restrict__ W,
                                                      const int*   __restrict__ labels,
                                                      float* __restrict__ xn,
                                                      float* __restrict__ cn) {
    const int row  = blockIdx.x * 8 + (threadIdx.x >> 5);
    const int lane = threadIdx.x & 31;

    // ---- x row ----
    const float* xr = x + (size_t)row * D_DIM;
    float4 vx[4];
    float ss = 0.f;
#pragma unroll
    for (int i = 0; i < 4; ++i) {
        vx[i] = *(const float4*)&xr[lane * 4 + i * 128];
        ss += vx[i].x * vx[i].x + vx[i].y * vx[i].y + vx[i].z * vx[i].z + vx[i].w * vx[i].w;
    }
#pragma unroll
    for (int m = 16; m >= 1; m >>= 1) ss += __shfl_xor(ss, m, 32);
    const float rnx = 1.f / fmaxf(sqrtf(ss), 1e-12f);
    float* xo = xn + (size_t)row * D_DIM;
#pragma unroll
    for (int i = 0; i < 4; ++i) {
        float4 o = make_float4(vx[i].x * rnx, vx[i].y * rnx, vx[i].z * rnx, vx[i].w * rnx);
        *(float4*)&xo[lane * 4 + i * 128] = o;
    }

    // ---- gathered center row ----
    const int idx = labels[row];
    const float* wr = W + (size_t)idx * D_DIM;
    float4 vc[4];
    float sc = 0.f;
#pragma unroll
    for (int i = 0; i < 4; ++i) {
        vc[i] = *(const float4*)&wr[lane * 4 + i * 128];
        sc += vc[i].x * vc[i].x + vc[i].y * vc[i].y + vc[i].z * vc[i].z + vc[i].w * vc[i].w;
    }
#pragma unroll
    for (int m = 16; m >= 1; m >>= 1) sc += __shfl_xor(sc, m, 32);
    const float rnc = 1.f / sqrtf(sc);
    float* co = cn + (size_t)row * D_DIM;
#pragma unroll
    for (int i = 0; i < 4; ++i) {
        float4 o = make_float4(vc[i].x * rnc, vc[i].y * rnc, vc[i].z * rnc, vc[i].w * rnc);
        *(float4*)&co[lane * 4 + i * 128] = o;
    }
}

// ---------------------------------------------------------------------------
// Kernel 2: fused 128x128-blocked GEMM (V_WMMA_F32_16X16X4_F32) + arcface
// diagonal margin + per-row partial reductions (sum exp(logit-64), sum logit).
// 8 waves per block; wave w owns a 16x128 strip (8 accumulators of v8f).
// Double-buffered K-panels staged with GLOBAL_LOAD_ASYNC_TO_LDS_B128.
// ---------------------------------------------------------------------------
__global__ __launch_bounds__(256) void gemm_arcface_reduce(const float* __restrict__ xn,
                                                           const float* __restrict__ cn,
                                                           float* __restrict__ sumexp_part,
                                                           float* __restrict__ sumlog_part,
                                                           float* __restrict__ diagv) {
    __shared__ float xsh[2][128 * LDSW];
    __shared__ float csh[2][128 * LDSW];

    const int bi = blockIdx.x, bj = blockIdx.y;
    const int t = threadIdx.x;
    const int wave = t >> 5, lane = t & 31;
    const int half = lane >> 4, lrow = lane & 15;
    const int i0 = bi * 128, j0 = bj * 128;

    v8f acc[8];
#pragma unroll
    for (int n = 0; n < 8; ++n)
#pragma unroll
        for (int e = 0; e < 8; ++e) acc[n][e] = 0.f;

    // per-thread staging slice: 4 float4 slots per panel
    const int q0 = t;                 // slot base; slots q0 + 256*it, it=0..3

    auto stage = [&](int kc, int buf) {
#pragma unroll
        for (int it = 0; it < 4; ++it) {
            const int q = q0 + 256 * it;          // 0..1023 float4 slots
            const int row = q >> 3;
            const int c4 = (q & 7) << 2;
#if USE_ASYNC_LDS
            __builtin_amdgcn_global_load_async_to_lds_b128(
                (gptr_v4i)&xn[(size_t)(i0 + row) * D_DIM + kc + c4],
                (lptr_v4i)&xsh[buf][row * LDSW + c4], 0, 0);
            __builtin_amdgcn_global_load_async_to_lds_b128(
                (gptr_v4i)&cn[(size_t)(j0 + row) * D_DIM + kc + c4],
                (lptr_v4i)&csh[buf][row * LDSW + c4], 0, 0);
#else
            // fallback: batched register staging
            float4 xv = *(const float4*)&xn[(size_t)(i0 + row) * D_DIM + kc + c4];
            float4 cv = *(const float4*)&cn[(size_t)(j0 + row) * D_DIM + kc + c4];
            *(float4*)&xsh[buf][row * LDSW + c4] = xv;
            *(float4*)&csh[buf][row * LDSW + c4] = cv;
#endif
        }
    };

    // prologue: fill buffer 0
    stage(0, 0);
#if USE_ASYNC_LDS
    __builtin_amdgcn_s_wait_asynccnt(0);
#endif
    __syncthreads();

    for (int kc = 0; kc < D_DIM; kc += 32) {
        const int cur = (kc >> 5) & 1;
        const int nxt = cur ^ 1;
        if (kc + 32 < D_DIM) stage(kc + 32, nxt);   // overlap next panel with compute

#pragma unroll
        for (int kk = 0; kk < 32; kk += 4) {
            // A fragment: 16x4, lane half selects K pair {kk,kk+1} vs {kk+2,kk+3}
            v2f a = *(const v2f*)&xsh[cur][(wave * 16 + lrow) * LDSW + kk + half * 2];
#pragma unroll
            for (int n = 0; n < 8; ++n) {
                v2f b = *(const v2f*)&csh[cur][(n * 16 + lrow) * LDSW + kk + half * 2];
                acc[n] = __builtin_amdgcn_wmma_f32_16x16x4_f32(
                    false, a, false, b, (short)0, acc[n], false, false);
            }
        }

#if USE_ASYNC_LDS
        __builtin_amdgcn_s_wait_asynccnt(0);        // next panel landed in LDS
#endif
        __syncthreads();                            // all waves done reading `cur`
    }

    // ---- epilogue: clip, diagonal margin, partial row reductions ----
    const bool diagBlock = (bi == bj);
    const int grow_base = i0 + wave * 16 + half * 8;   // + e
    float sumexp[8], sumlog[8];
#pragma unroll
    for (int e = 0; e < 8; ++e) { sumexp[e] = 0.f; sumlog[e] = 0.f; }

#pragma unroll
    for (int n = 0; n < 8; ++n) {
#pragma unroll
        for (int e = 0; e < 8; ++e) {
            float v = fminf(fmaxf(acc[n][e], -1.f), 1.f);
            const bool isDiag = diagBlock && (n == wave) && (lrow == half * 8 + e);
            if (isDiag) v = cosf(acosf(v) + 0.5f);     // arcface margin M=0.5
            const float logit = 64.f * v;              // scale S=64
            if (isDiag) diagv[grow_base + e] = logit;
            sumlog[e] += logit;
            sumexp[e] += __expf(logit - 64.f);         // logit <= 64 -> no overflow
        }
    }

    // reduce across the 16 lanes of each half (xor masks stay within half)
#pragma unroll
    for (int e = 0; e < 8; ++e) {
        float se = sumexp[e], sl = sumlog[e];
#pragma unroll
        for (int m = 8; m >= 1; m >>= 1) {
            se += __shfl_xor(se, m, 32);
            sl += __shfl_xor(sl, m, 32);
        }
        if (lrow == 0) {
            const int grow = grow_base + e;
            sumexp_part[(size_t)bj * B_ROWS + grow] = se;
            sumlog_part[(size_t)bj * B_ROWS + grow] = sl;
        }
    }
}

// ---------------------------------------------------------------------------
// Kernel 3: deterministic final reduction -> scalar loss
// loss = (1/B) * sum_i [ (1-eps)*(lse_i - diag_i) + (eps/C)*(B*lse_i - sumlog_i) ]
// ---------------------------------------------------------------------------
__global__ __launch_bounds__(256) void final_reduce(const float* __restrict__ sumexp_part,
                                                    const float* __restrict__ sumlog_part,
                                                    const float* __restrict__ diagv,
                                                    float* __restrict__ out) {
    __shared__ float red[256];
    const float one_m_eps = 0.9f;
    const float eps_over_c = 0.1f / 500000.f;
    float acc = 0.f;
    for (int row = threadIdx.x; row < B_ROWS; row += 256) {
        float se = 0.f, sl = 0.f;
        for (int jb = 0; jb < NJB; ++jb) {           // fixed order -> deterministic
            se += sumexp_part[(size_t)jb * B_ROWS + row];
            sl += sumlog_part[(size_t)jb * B_ROWS + row];
        }
        const float lse = 64.f + logf(se);
        const float d = diagv[row];
        acc += one_m_eps * (lse - d) + eps_over_c * ((float)B_ROWS * lse - sl);
    }
    red[threadIdx.x] = acc;
    __syncthreads();
    for (int s = 128; s > 0; s >>= 1) {
        if (threadIdx.x < s) red[threadIdx.x] += red[threadIdx.x + s];
        __syncthreads();
    }
    if (threadIdx.x == 0) out[0] = red[0] / (float)B_ROWS;
}

// ---------------------------------------------------------------------------
extern "C" void kernel_launch(void* const* d_in, const int* in_sizes, int n_in,
                              void* d_out, int out_size, void* d_ws, size_t ws_size,
                              hipStream_t stream) {
    const float* x      = (const float*)d_in[0];
    const int*   labels = (const int*)d_in[1];
    const float* W      = (const float*)d_in[2];
    float* out = (float*)d_out;

    float* ws = (float*)d_ws;
    float* xn          = ws;                                   // 8192*512
    float* cn          = xn + (size_t)B_ROWS * D_DIM;          // 8192*512
    float* sumexp_part = cn + (size_t)B_ROWS * D_DIM;          // 64*8192
    float* sumlog_part = sumexp_part + (size_t)NJB * B_ROWS;   // 64*8192
    float* diagv       = sumlog_part + (size_t)NJB * B_ROWS;   // 8192

    normalize_rows<<<B_ROWS / 8, 256, 0, stream>>>(x, W, labels, xn, cn);

    dim3 grid(NJB, NJB);
    gemm_arcface_reduce<<<grid, 256, 0, stream>>>(xn, cn, sumexp_part, sumlog_part, diagv);

    final_reduce<<<1, 256, 0, stream>>>(sumexp_part, sumlog_part, diagv, out);
}